// smooth_Loss_43989055045797
// MI455X (gfx1250) — compile-verified
//
#include <hip/hip_runtime.h>

typedef float v2f __attribute__((ext_vector_type(2)));
typedef float v8f __attribute__((ext_vector_type(8)));
typedef unsigned int v4u __attribute__((ext_vector_type(4)));
typedef int v4i __attribute__((ext_vector_type(4)));
typedef int v8i __attribute__((ext_vector_type(8)));

#define PLANE_ELEMS 4096   // 64*64 floats per plane (16 KB)

__global__ void tv_zero_kernel(float* out) {
    out[0] = 0.0f;
}

// One workgroup per 64x64 plane; 256 threads = 8 wave32 waves.
// TDM (tensor_load_to_lds) DMAs the 16KB plane into LDS in a single
// instruction issued by wave 0; stencil is computed from LDS with
// register-carried vertical neighbors; wave reduction via WMMA;
// one global f32 atomic per block.
__global__ __launch_bounds__(256) void tv_plane_kernel(const float* __restrict__ fai,
                                                       float* __restrict__ out) {
    __shared__ struct {
        float tile[PLANE_ELEMS];   // LDS offset 0 (first/only shared object)
        float acc;
    } sh;

    const int t = threadIdx.x;
    if (t == 0) sh.acc = 0.0f;

    const unsigned long long gaddr =
        (unsigned long long)(const void*)(fai + (long long)blockIdx.x * PLANE_ELEMS);

#if __has_builtin(__builtin_amdgcn_tensor_load_to_lds) && __has_builtin(__builtin_amdgcn_s_wait_tensorcnt)
    if (t < 32) {
        // ---- Tensor DMA descriptor (D#), 1-D tile: 4096 x f32 ----
        v4u g0;
        g0[0] = 1u;                                        // count=1 (valid), user mode, no gather
        g0[1] = 0u;                                        // lds_addr = 0 (sh.tile)
        g0[2] = (unsigned)(gaddr & 0xFFFFFFFFu);           // global_addr[31:0]
        g0[3] = (unsigned)((gaddr >> 32) & 0x01FFFFFFu)    // global_addr[56:32]
                | 0x80000000u;                             // type = 2 ("image")
        v8i g1;
        g1[0] = (int)(2u << 16);                           // workgroup_mask=0, data_size=2 (4B)
        g1[1] = (int)((PLANE_ELEMS & 0xFFFF) << 16);       // tensor_dim0[15:0] = 4096
        g1[2] = (int)((PLANE_ELEMS >> 16) | (1u << 16));   // tensor_dim0[31:16], tensor_dim1 = 1
        g1[3] = (int)((unsigned)PLANE_ELEMS << 16);        // tile_dim0 = 4096
        g1[4] = 1;                                         // tile_dim1 = 1, tile_dim2 = 0
        g1[5] = PLANE_ELEMS;                               // tensor_dim0_stride[31:0] = 4096
        g1[6] = (int)((unsigned)PLANE_ELEMS << 16);        // tensor_dim1_stride[15:0] = 4096
        g1[7] = 0;                                         // tensor_dim1_stride[47:16] = 0
        v4i g2 = {0, 0, 0, 0};
        v4i g3 = {0, 0, 0, 0};
        v8i g4 = {0, 0, 0, 0, 0, 0, 0, 0};                 // extra group (clang-23 6-arg form)
        __builtin_amdgcn_tensor_load_to_lds(g0, g1, g2, g3, g4, 0);
        __builtin_amdgcn_s_wait_tensorcnt(0);              // TDM done before we signal barrier
    }
#else
    {
        const float4* __restrict__ src =
            (const float4*)(fai + (long long)blockIdx.x * PLANE_ELEMS);
        float4* dst = (float4*)sh.tile;
#pragma unroll
        for (int i = 0; i < 4; ++i) dst[t + i * 256] = src[t + i * 256];
    }
#endif
    __syncthreads();

    // ---- Stencil: column-per-thread, register-carried vertical neighbors ----
    // 64 columns x 4 row-bands; rows 1..62 interior, cols 1..62 interior.
    const int c = t & 63;
    const int q = t >> 6;                  // wave-uniform (2 waves per band)
    float partial = 0.0f;
    if (c >= 1 && c <= 62) {
        const int rbeg = 1 + 16 * q;
        const int rend = (q == 3) ? 62 : (16 * q + 16);
        float prev = sh.tile[(rbeg - 1) * 64 + c];
        float cur  = sh.tile[rbeg * 64 + c];
        for (int r = rbeg; r <= rend; ++r) {
            const float nxt = sh.tile[(r + 1) * 64 + c];
            const float lf  = sh.tile[r * 64 + c - 1];
            const float rt  = sh.tile[r * 64 + c + 1];
            partial += fabsf(cur - prev) + fabsf(cur - nxt)
                     + fabsf(cur - lf)   + fabsf(cur - rt);
            prev = cur;
            cur  = nxt;
        }
    }

    // ---- Wave-level reduction via V_WMMA_F32_16X16X4_F32 (EXEC all-ones here) ----
    // A: lane L -> A[L%16][2*(L/16)] = partial; B = ones.
    // D row-sums: lane-local sum of the 8 D components on lanes N==0 (lanes 0,16)
    // gives complementary halves of the wave total.
#if __has_builtin(__builtin_amdgcn_wmma_f32_16x16x4_f32)
    {
        v2f a; a.x = partial; a.y = 0.0f;
        v2f b; b.x = 1.0f;    b.y = 1.0f;
        v8f creg = {};
        v8f d = __builtin_amdgcn_wmma_f32_16x16x4_f32(
            false, a, false, b, (short)0, creg, false, false);
        float s = d[0] + d[1] + d[2] + d[3] + d[4] + d[5] + d[6] + d[7];
        if ((t & 15) == 0) {
            atomicAdd(&sh.acc, s);         // ds_add_f32
        }
    }
#else
    {
        float s = partial;
        for (int off = 16; off > 0; off >>= 1)
            s += __shfl_down(s, off, 32);
        if ((t & 31) == 0)
            atomicAdd(&sh.acc, s);
    }
#endif

    __syncthreads();
    if (t == 0) {
        atomicAdd(out, sh.acc);            // global_atomic_add_f32, 1 per plane
    }
}

extern "C" void kernel_launch(void* const* d_in, const int* in_sizes, int n_in,
                              void* d_out, int out_size, void* d_ws, size_t ws_size,
                              hipStream_t stream) {
    (void)n_in; (void)out_size; (void)d_ws; (void)ws_size;

    const float* fai = (const float*)d_in[0];
    float* out = (float*)d_out;

    const int total_elems = in_sizes[0];            // 4096*2*64*64
    const int planes = total_elems / PLANE_ELEMS;   // 8192

    tv_zero_kernel<<<1, 1, 0, stream>>>(out);
    tv_plane_kernel<<<planes, 256, 0, stream>>>(fai, out);
}